// SpecialLayer_69630009803300
// MI455X (gfx1250) — compile-verified
//
#include <hip/hip_runtime.h>

// SpecialLayer on MI455X (gfx1250):
//   out[i, :] = prod_{j in activated, relu(x[i,j]) != 0} relu(x[i,j])   (broadcast)
//   zeroed globally if every activated relu value over the whole batch is zero.
//
// Memory-bound: ~24 MB mandatory column reads + 128 MB streamed output writes
// => ~6.5 us at 23.3 TB/s. No matrix op is appropriate (would inflate reads 5x
// and add 34 GFLOP of dead work), so this kernel intentionally emits no WMMA;
// the CDNA5-specific paths used are wave32 ballot reduction and non-temporal
// (TH_STORE_NT) b128 streaming stores for the write-once output.

typedef float v4f __attribute__((ext_vector_type(4)));

#define SL_F 512  // IN_FEATURES (fixed by the reference)

// d_ws layout in 4-byte units:
//   wsi[0]            : global nonzero flag
//   wsi[1]            : nact (number of activated diagonal entries)
//   wsi[2   .. 2+F)   : activated column indices (ascending order)
//   wsf[2+F .. 2+2F)  : diagonal values at those columns
#define WS_IDX   2
#define WS_DVAL  (2 + SL_F)

// ---- Pass 0: one thread scans the diagonal, builds the compact index list
// (in ascending index order so the fp product order matches the reference),
// and resets the flag. Runs fresh every call -> deterministic.
__global__ void sl_build_idx_kernel(const float* __restrict__ diag,
                                    int* wsi, float* wsf) {
    if (blockIdx.x == 0 && threadIdx.x == 0) {
        int n = 0;
        for (int j = 0; j < SL_F; ++j) {
            float dv = diag[(size_t)j * SL_F + j];
            if (dv != 0.0f) {
                wsi[WS_IDX + n]  = j;
                wsf[WS_DVAL + n] = dv;
                ++n;
            }
        }
        wsi[1] = n;
        wsi[0] = 0;
    }
}

// ---- Pass 1: global "any nonzero" flag. One thread per row, <=5 gathered
// loads (3 cache lines/row), wave32 ballot reduction, one atomicOr per wave.
__global__ void sl_flag_kernel(const float* __restrict__ x,
                               int* wsi, float* wsf, int B) {
    const int row  = blockIdx.x * blockDim.x + threadIdx.x;
    const int nact = wsi[1];
    bool nz = false;
    if (row < B) {
        const float* xr = x + (size_t)row * SL_F;
        for (int k = 0; k < nact; ++k) {
            const int   j  = wsi[WS_IDX + k];
            const float v  = fmaxf(xr[j], 0.0f) * wsf[WS_DVAL + k];
            nz |= (v != 0.0f);
        }
    }
    // wave32 reduction: one atomic per wave at most
    if (__any((int)nz) && (threadIdx.x & (warpSize - 1)) == 0) {
        atomicOr(&wsi[0], 1);
    }
}

// ---- Pass 2: broadcast writer. 256 threads / block handle 2 rows
// (2 x 512 floats = 2 x 128 float4 stores). Lane 0 of each row recomputes the
// <=5-factor product (column lines are L2-resident from pass 1), shares it via
// LDS, and every lane issues a non-temporal 128-bit store (write-once stream,
// keep it out of L2's way).
__global__ void sl_out_kernel(const float* __restrict__ x,
                              const int* wsi, const float* wsf,
                              float* __restrict__ out, int B) {
    __shared__ float sprod[2];
    const int t    = threadIdx.x;
    const int rib  = t >> 7;        // row-in-block: 0 or 1
    const int lane = t & 127;       // float4 slot within the row
    const int row  = blockIdx.x * 2 + rib;

    const int flag = wsi[0];
    const int nact = wsi[1];

    if (lane == 0 && row < B) {
        const float* xr = x + (size_t)row * SL_F;
        float prod = 1.0f;
        for (int k = 0; k < nact; ++k) {   // ascending index order == reference
            const int   j = wsi[WS_IDX + k];
            const float v = fmaxf(xr[j], 0.0f) * wsf[WS_DVAL + k];
            if (v != 0.0f) prod *= v;
        }
        sprod[rib] = prod;
    }
    __syncthreads();

    if (row < B) {
        const float p = flag ? sprod[rib] : 0.0f;
        v4f vec = {p, p, p, p};
        __builtin_nontemporal_store(
            vec, (v4f*)(out + (size_t)row * SL_F + (size_t)lane * 4));
    }
}

extern "C" void kernel_launch(void* const* d_in, const int* in_sizes, int n_in,
                              void* d_out, int out_size, void* d_ws, size_t ws_size,
                              hipStream_t stream) {
    const float* x    = (const float*)d_in[0];  // [B, 512] f32
    const float* diag = (const float*)d_in[1];  // [512, 512] f32 (diagonal mask)
    float* out = (float*)d_out;                 // [B, 512] f32

    const int B = in_sizes[0] / SL_F;           // 65536

    int*   wsi = (int*)d_ws;
    float* wsf = (float*)d_ws;

    sl_build_idx_kernel<<<1, 1, 0, stream>>>(diag, wsi, wsf);
    sl_flag_kernel<<<(B + 255) / 256, 256, 0, stream>>>(x, wsi, wsf, B);
    sl_out_kernel<<<(B + 1) / 2, 256, 0, stream>>>(x, wsi, wsf, out, B);
}